// LDStack_41927470744181
// MI455X (gfx1250) — compile-verified
//
#include <hip/hip_runtime.h>

typedef __attribute__((ext_vector_type(2))) float v2f;
typedef __attribute__((ext_vector_type(8))) float v8f;

// Problem constants
constexpr int TT  = 2048;   // T
constexpr int DD  = 128;    // d
constexpr int KH  = 32;     // k
constexpr int NH  = 32;     // n
constexpr int KN  = KH * NH;   // 1024
constexpr int MD  = 128;    // m
constexpr int LT  = 16;     // time-tile
constexpr int NCH = TT / LT;   // 128 chunks
constexpr int KTOT = 2 * KN;   // 2048 GEMM inner dim (Sr|Si)
constexpr int NCHK = KTOT / 64; // 32 K-chunks in main GEMM

// LDS strides (bank-conflict padding)
constexpr int SSTR = KTOT + 4;   // 2052, A-frag rows hit distinct banks
constexpr int BSTR = MD + 8;     // 136, B-frag halves hit disjoint banks

// ---------------------------------------------------------------- prep: lam, Bp
__global__ __launch_bounds__(1024) void k_prep(const float* __restrict__ theta,
                                               float* __restrict__ lamR, float* __restrict__ lamI,
                                               float* __restrict__ BpR,  float* __restrict__ BpI) {
  int kn = threadIdx.x;            // 0..1023
  int kk = kn >> 5, j = kn & 31;
  float thj = (j < 16) ? theta[kk * 16 + j] : -theta[kk * 16 + (j - 16)];
  lamR[kn] = cosf(thj);
  lamI[kn] = sinf(thj);
  // lnBp[j] = -sum_{i!=j} log(1 - exp(i(th_i - th_j)))   (complex log)
  float ar = 0.f, ai = 0.f;
  for (int i = 0; i < 32; ++i) {
    if (i == j) continue;
    float thi = (i < 16) ? theta[kk * 16 + i] : -theta[kk * 16 + (i - 16)];
    float d = thi - thj;
    float cd = cosf(d), sd = sinf(d);
    float zr = 1.0f - cd, zi = -sd;            // 1 - lam_i/lam_j
    ar -= 0.5f * logf(zr * zr + zi * zi);
    ai -= atan2f(zi, zr);
  }
  float eb = expf(ar);
  BpR[kn] = eb * cosf(ai);
  BpI[kn] = eb * sinf(ai);
}

// --------------------------------------------- makeB: Bmat = [CpR ; -CpI]  (2048 x 128)
__global__ __launch_bounds__(256) void k_makeB(const float* __restrict__ lnCr,
                                               const float* __restrict__ lnCi,
                                               float* __restrict__ Bmat) {
  int e = blockIdx.x * 256 + threadIdx.x;      // 0 .. 262143
  int row = e >> 7;
  if (row < KN) {
    float mg = expf(lnCr[e]);
    Bmat[e] = mg * cosf(lnCi[e]);
  } else {
    int e2 = e - KN * MD;
    float mg = expf(lnCr[e2]);
    Bmat[e] = -mg * sinf(lnCi[e2]);
  }
}

// ------------------------------------------------------- proj: xp = x @ R  via WMMA f32
__global__ __launch_bounds__(64) void k_proj(const float* __restrict__ x,
                                             const float* __restrict__ R,
                                             float* __restrict__ xp) {
  int blk  = blockIdx.x;                 // row tile (16 rows each), 1024 tiles
  int tid  = threadIdx.x;
  int wave = tid >> 5;                   // 2 waves -> n-tiles 0..15 / 16..31
  int lane = tid & 31;
  int half = lane >> 4, l15 = lane & 15;
  int n0   = wave * 16;
  int row  = blk * 16 + l15;             // A-matrix M row for this lane

  v8f acc = {0.f, 0.f, 0.f, 0.f, 0.f, 0.f, 0.f, 0.f};
#pragma unroll
  for (int kk = 0; kk < DD; kk += 4) {
    v2f a, b;
    const float* ap = x + row * DD + kk + half * 2;
    a.x = ap[0];
    a.y = ap[1];
    const float* bp = R + (kk + half * 2) * KH + n0 + l15;
    b.x = bp[0];
    b.y = bp[KH];
    acc = __builtin_amdgcn_wmma_f32_16x16x4_f32(false, a, false, b, (short)0, acc,
                                                false, false);
  }
#pragma unroll
  for (int r = 0; r < 8; ++r) {
    int M = r + half * 8;
    xp[(blk * 16 + M) * KH + n0 + l15] = acc[r];
  }
}

// ------------------------------------- chunkscan: V_c = sum_j lam^{15-j} * Bp * xp[cL+j]
__global__ __launch_bounds__(1024) void k_chunkscan(const float* __restrict__ xp,
                                                    const float* __restrict__ lamR,
                                                    const float* __restrict__ lamI,
                                                    const float* __restrict__ BpR,
                                                    const float* __restrict__ BpI,
                                                    float* __restrict__ VR,
                                                    float* __restrict__ VI) {
  int blk = blockIdx.x;                  // b*128 + c
  int bb = blk >> 7, c = blk & 127;
  int kn = threadIdx.x;
  int kk = kn >> 5;
  float lr = lamR[kn], li = lamI[kn], br = BpR[kn], bi = BpI[kn];
  float sr = 0.f, si = 0.f;
  const float* xpp = xp + (bb * TT + c * LT) * KH + kk;
#pragma unroll
  for (int j = 0; j < LT; ++j) {
    float xv = xpp[j * KH];              // all lanes in wave share this address
    float nr = lr * sr - li * si + br * xv;
    float ni = lr * si + li * sr + bi * xv;
    sr = nr; si = ni;
  }
  VR[(bb * NCH + c) * KN + kn] = sr;
  VI[(bb * NCH + c) * KN + kn] = si;
}

// -------------------- carryprefix: u_end[c] = lam^16 * u_end[c-1] + V_c  (in place)
__global__ __launch_bounds__(1024) void k_carry(const float* __restrict__ theta,
                                                float* __restrict__ VR,
                                                float* __restrict__ VI) {
  int bb = blockIdx.x;                   // 0..7
  int kn = threadIdx.x;
  int kk = kn >> 5, nn = kn & 31;
  float th = (nn < 16) ? theta[kk * 16 + nn] : -theta[kk * 16 + (nn - 16)];
  float a = 16.0f * th;                  // lam^16 exactly, no drift
  float lr = cosf(a), li = sinf(a);
  float ur = 0.f, ui = 0.f;
  float* pr = VR + bb * NCH * KN + kn;
  float* pi = VI + bb * NCH * KN + kn;
  for (int c = 0; c < NCH; ++c) {
    float vr = pr[c * KN], vi = pi[c * KN];
    float nr = lr * ur - li * ui + vr;
    float ni = lr * ui + li * ur + vi;
    ur = nr; ui = ni;
    pr[c * KN] = ur;                     // carry[c] = u at end of chunk c
    pi[c * KN] = ui;
  }
}

// CDNA5 async LDS copy: L2 -> LDS with no VGPR round trip, tracked by ASYNCcnt.
__device__ __forceinline__ void async_b128(unsigned int lds_addr, unsigned int voff,
                                           const float* sbase) {
  asm volatile("global_load_async_to_lds_b128 %0, %1, %2"
               :: "v"(lds_addr), "v"(voff), "s"(sbase)
               : "memory");
}
__device__ __forceinline__ void wait_async0() {
  asm volatile("s_wait_asynccnt 0x0" ::: "memory");
}

// ------------------------------- main: local scan replay -> LDS, WMMA GEMM, epilogue
__global__ __launch_bounds__(256) void k_main(const float* __restrict__ xp,
                                              const float* __restrict__ lamR,
                                              const float* __restrict__ lamI,
                                              const float* __restrict__ BpR,
                                              const float* __restrict__ BpI,
                                              const float* __restrict__ carR,
                                              const float* __restrict__ carI,
                                              const float* __restrict__ Bmat,
                                              const float* __restrict__ Dm,
                                              const float* __restrict__ Dov,
                                              float* __restrict__ out) {
  extern __shared__ float smem[];
  float* Sbuf = smem;                    // 16 x SSTR  (t-major; cols 0..1023=Sr, 1024..2047=Si)
  float* xt   = Sbuf + LT * SSTR;        // 16 x 32
  float* Bb0  = xt + LT * KH;            // 64 x BSTR  (double-buffered B chunk)
  float* Bb1  = Bb0 + 64 * BSTR;         // 64 x BSTR

  const int tid = threadIdx.x;
  const int blk = blockIdx.x;            // b*128 + c
  const int bb = blk >> 7;
  const int c  = blk & 127;
  const int t0 = c * LT;

  const unsigned int lds_b0 = (unsigned int)(size_t)Bb0;
  const unsigned int lds_b1 = (unsigned int)(size_t)Bb1;

  // kick off async DMA of the first B chunk (overlaps with the scan replay below)
  {
#pragma unroll
    for (int it = 0; it < 8; ++it) {
      int i4 = tid + it * 256;           // 0..2047 float4s of a 64x128 chunk
      int row = i4 >> 5;
      int c4 = i4 & 31;
      async_b128(lds_b0 + (unsigned int)((row * BSTR + c4 * 4) * 4),
                 (unsigned int)((row * MD + c4 * 4) * 4), Bmat);
    }
  }

  // stage xp tile (16 x 32, contiguous)
  for (int e = tid; e < LT * KH; e += 256)
    xt[e] = xp[(bb * TT + t0) * KH + e];
  __syncthreads();

  // local scan replay: 4 kn per thread, kn = q*256 + tid (coalesced / conflict-free)
  float sr[4], si[4], lr[4], li[4], br[4], bi[4];
  int kidx[4];
#pragma unroll
  for (int q = 0; q < 4; ++q) {
    int kn = q * 256 + tid;
    kidx[q] = kn >> 5;
    lr[q] = lamR[kn]; li[q] = lamI[kn];
    br[q] = BpR[kn];  bi[q] = BpI[kn];
    if (c > 0) {
      sr[q] = carR[(bb * NCH + (c - 1)) * KN + kn];
      si[q] = carI[(bb * NCH + (c - 1)) * KN + kn];
    } else {
      sr[q] = 0.f; si[q] = 0.f;
    }
  }
#pragma unroll
  for (int i = 0; i < LT; ++i) {
#pragma unroll
    for (int q = 0; q < 4; ++q) {
      int kn = q * 256 + tid;
      Sbuf[i * SSTR + kn]      = sr[q];          // S_t = u_{t-1} (shifted state)
      Sbuf[i * SSTR + KN + kn] = si[q];
      float xv = xt[i * KH + kidx[q]];
      float nr = lr[q] * sr[q] - li[q] * si[q] + br[q] * xv;
      float ni = lr[q] * si[q] + li[q] * sr[q] + bi[q] * xv;
      sr[q] = nr; si[q] = ni;
    }
  }
  wait_async0();        // chunk 0 DMA complete (this wave)
  __syncthreads();      // all waves: Sbuf + chunk 0 ready

  // GEMM: (16 x 2048) * (2048 x 128), each wave owns one 16-col m-tile.
  // Double-buffered: async-DMA chunk ch+1 while WMMAs consume chunk ch.
  const int wave = tid >> 5;
  const int lane = tid & 31;
  const int half = lane >> 4;
  const int l15  = lane & 15;
  const int col0 = wave * 16;

  v8f acc = {0.f, 0.f, 0.f, 0.f, 0.f, 0.f, 0.f, 0.f};
  for (int ch = 0; ch < NCHK; ++ch) {
    int kk0 = ch * 64;
    if (ch + 1 < NCHK) {
      // target buffer was last read in iteration ch-1; the barrier ending ch-1
      // guarantees all waves are done with it.
      unsigned int dst = ((ch + 1) & 1) ? lds_b1 : lds_b0;
      int src0 = (kk0 + 64) * MD;
#pragma unroll
      for (int it = 0; it < 8; ++it) {
        int i4 = tid + it * 256;
        int row = i4 >> 5;
        int c4 = i4 & 31;
        async_b128(dst + (unsigned int)((row * BSTR + c4 * 4) * 4),
                   (unsigned int)((src0 + row * MD + c4 * 4) * 4), Bmat);
      }
    }
    const float* Bc = (ch & 1) ? Bb1 : Bb0;
#pragma unroll
    for (int kk = 0; kk < 64; kk += 4) {
      v2f a = *(const v2f*)(Sbuf + l15 * SSTR + kk0 + kk + half * 2);
      v2f b;
      const float* bp = Bc + (kk + half * 2) * BSTR + col0 + l15;
      b.x = bp[0];
      b.y = bp[BSTR];
      acc = __builtin_amdgcn_wmma_f32_16x16x4_f32(false, a, false, b, (short)0, acc,
                                                  false, false);
    }
    wait_async0();      // my chunk-(ch+1) loads done
    __syncthreads();    // everyone's done; next buffer valid, current buffer free
  }

  // epilogue: + sum_k xp*D, /32, + Do
  int m = col0 + l15;
  float dsum[8] = {0.f, 0.f, 0.f, 0.f, 0.f, 0.f, 0.f, 0.f};
  for (int kq = 0; kq < KH; ++kq) {
    float dv = Dm[kq * MD + m];
#pragma unroll
    for (int r = 0; r < 8; ++r) {
      int M = r + half * 8;
      dsum[r] += xt[M * KH + kq] * dv;
    }
  }
  float dov = Dov[m];
#pragma unroll
  for (int r = 0; r < 8; ++r) {
    int M = r + half * 8;
    out[(bb * TT + t0 + M) * MD + m] = (acc[r] + dsum[r]) * (1.0f / 32.0f) + dov;
  }
}

// ------------------------------------------------------------------------- launch
extern "C" void kernel_launch(void* const* d_in, const int* in_sizes, int n_in,
                              void* d_out, int out_size, void* d_ws, size_t ws_size,
                              hipStream_t stream) {
  const float* x     = (const float*)d_in[0];   // [8,2048,128]
  const float* R     = (const float*)d_in[1];   // [128,32]
  const float* theta = (const float*)d_in[2];   // [32,16]
  const float* lnCr  = (const float*)d_in[3];   // [32,32,128]
  const float* lnCi  = (const float*)d_in[4];   // [32,32,128]
  const float* Dm    = (const float*)d_in[5];   // [32,128]
  const float* Dov   = (const float*)d_in[6];   // [128]
  float* out = (float*)d_out;                   // [8,2048,128]

  float* ws   = (float*)d_ws;
  float* lamR = ws;                             // 1024
  float* lamI = lamR + KN;                      // 1024
  float* BpR  = lamI + KN;                      // 1024
  float* BpI  = BpR + KN;                       // 1024
  float* Bmat = BpI + KN;                       // 2048*128
  float* xp   = Bmat + KTOT * MD;               // 16384*32
  float* VR   = xp + (8 * TT) * KH;             // 8*128*1024
  float* VI   = VR + 8 * NCH * KN;              // 8*128*1024

  k_prep<<<1, 1024, 0, stream>>>(theta, lamR, lamI, BpR, BpI);
  k_makeB<<<(KTOT * MD) / 256, 256, 0, stream>>>(lnCr, lnCi, Bmat);
  k_proj<<<(8 * TT) / 16, 64, 0, stream>>>(x, R, xp);
  k_chunkscan<<<8 * NCH, 1024, 0, stream>>>(xp, lamR, lamI, BpR, BpI, VR, VI);
  k_carry<<<8, 1024, 0, stream>>>(theta, VR, VI);

  size_t shmem = (size_t)(LT * SSTR + LT * KH + 2 * 64 * BSTR) * sizeof(float);
  k_main<<<8 * NCH, 256, shmem, stream>>>(xp, lamR, lamI, BpR, BpI, VR, VI,
                                          Bmat, Dm, Dov, out);
}